// Triplet_SAC_7438883356860
// MI455X (gfx1250) — compile-verified
//
#include <hip/hip_runtime.h>
#include <float.h>
#include <math.h>

// Problem constants (from reference)
#define N      4096
#define D      512
#define NCLS   1024
#define EPS    1e-12f
#define ALPHA  1.0f
#define GAMMA  0.5f
#define THETA  0.1f
#define MARGIN 0.0f

// Tiling
#define BM   64             // i-rows per workgroup (resident in LDS, full D)
#define BN   128            // j-cols per step (4 wave-groups x 2 subtiles x 16)
#define KC   128            // K chunk for the B stage (double buffered)
#define NKC  (D / KC)       // 4 chunks per j-step
#define LDA  (D + 4)        // A row stride: TDM pads 2 dwords per 256 dwords -> 516
#define LDB  (KC + 4)       // B row stride: TDM pads 4 dwords per 128 dwords -> 132
#define TCH  ((N / BN) * NKC)   // total B chunks per block
#define FROB_BLOCKS 256

typedef float v2f  __attribute__((ext_vector_type(2)));
typedef float v8f  __attribute__((ext_vector_type(8)));
typedef unsigned int u32x4 __attribute__((ext_vector_type(4)));
typedef unsigned int u32x8 __attribute__((ext_vector_type(8)));

// LDS byte-offset of a generic pointer into shared memory (ptrtoint of AS(3)).
typedef __attribute__((address_space(3))) float lds_float_t;
static __device__ __forceinline__ unsigned lds_off(const float* p) {
    return (unsigned)(unsigned long long)(lds_float_t*)p;
}

// TDM: issue tensor_load_to_lds for a 2-D tile of f32.
//  rows x cols elements, row stride 'stride' elements, LDS rows padded per (pi,pa) codes.
//  pad_interval code pi: pad after 2^(pi+1) dwords; pad_amount code pa: pa+1 dwords.
static __device__ __forceinline__ void tdm_load_2d(unsigned lds_byte_off,
                                                   const float* gptr,
                                                   unsigned cols, unsigned rows,
                                                   unsigned stride,
                                                   unsigned pi, unsigned pa) {
    const unsigned long long ga = (unsigned long long)gptr;
    u32x4 g0;
    g0[0] = 1u;                                            // count=1, user descriptor
    g0[1] = lds_byte_off;                                  // lds_addr (bytes)
    g0[2] = (unsigned)(ga & 0xFFFFFFFFull);                // global_addr[31:0]
    g0[3] = (unsigned)((ga >> 32) & 0x1FFFFFFull)          // global_addr[56:32]
          | (2u << 30);                                    // type = 2 ("image")
    u32x8 g1;
    g1[0] = (2u << 16)                                     // data_size = 4 bytes
          | (1u << 20)                                     // pad_enable
          | (pi << 22) | (pa << 25);                       // pad interval / amount codes
    g1[1] = (cols & 0xFFFFu) << 16;                        // tensor_dim0[15:0]
    g1[2] = (cols >> 16) | ((rows & 0xFFFFu) << 16);       // tensor_dim0[31:16] | tensor_dim1[15:0]
    g1[3] = (rows >> 16) | (cols << 16);                   // tensor_dim1[31:16] | tile_dim0
    g1[4] = rows;                                          // tile_dim1 | tile_dim2(=0)<<16
    g1[5] = stride;                                        // tensor_dim0_stride[31:0]
    g1[6] = 0u;                                            // stride[47:32] | dim1_stride lo
    g1[7] = 0u;
    asm volatile("tensor_load_to_lds %0, %1" :: "s"(g0), "s"(g1) : "memory");
}

// ---------------------------------------------------------------- row squared norms
__global__ void rowsq_kernel(const float* __restrict__ X, float* __restrict__ row_sq) {
    const int wave = threadIdx.x >> 5, lane = threadIdx.x & 31;
    const int row  = blockIdx.x * 8 + wave;
    const float* xr = X + (size_t)row * D;
    float s = 0.f;
    for (int k = lane; k < D; k += 32) { float v = xr[k]; s += v * v; }
    #pragma unroll
    for (int m = 16; m; m >>= 1) s += __shfl_xor(s, m, 32);
    if (lane == 0) row_sq[row] = s;
}

// ---------------------------------------------------------------- fused GEMM + hard mining
// grid = N/BM blocks of 256 threads (8 waves).
// Wave w: ipair = w&1 -> i-tiles {2*ipair, 2*ipair+1}; jgroup = w>>1 -> 2 j-subtiles.
// A staged once by TDM; B chunks double-buffered by TDM, DMA(t+1) overlapped with compute(t).
__global__ void __launch_bounds__(256)
tripdist_kernel(const float* __restrict__ X, const int* __restrict__ tgt,
                const float* __restrict__ row_sq, float* __restrict__ partial) {
    extern __shared__ float smem[];
    float* As  = smem;                       // BM x LDA   (TDM pad layout: +2 dw gap @256)
    float* Bsb = As + BM * LDA;              // 2 x BN x LDB double buffer
    float* sqA = Bsb + 2 * BN * LDB;         // BM
    int*   tA  = (int*)(sqA + BM);           // BM
    float* apJ = (float*)(tA + BM);          // 4 x BM
    float* anJ = apJ + 4 * BM;               // 4 x BM
    float* red = anJ + 4 * BM;               // BM

    const int tid  = threadIdx.x;
    const int wave = tid >> 5, lane = tid & 31;
    const int i0   = blockIdx.x * BM;

    const unsigned asOff = lds_off(As);
    const unsigned bsOff = lds_off(Bsb);
    const unsigned bufBytes = (unsigned)(BN * LDB * sizeof(float));

    if (wave == 0) {
        // A block: 64 rows x 512 cols, pad 2 dwords per 256 dwords -> row stride 516.
        tdm_load_2d(asOff, X + (size_t)i0 * D, D, BM, D, /*pi=*/7u, /*pa=*/1u);
        // First B chunk: 128 rows x 128 cols, pad 4 dwords per 128 dwords -> stride 132.
        tdm_load_2d(bsOff, X, KC, BN, D, /*pi=*/6u, /*pa=*/3u);
    }
    if (tid < BM) { sqA[tid] = row_sq[i0 + tid]; tA[tid] = tgt[i0 + tid]; }
    __syncthreads();

    const int ipair  = wave & 1;     // i-tile pair
    const int jgroup = wave >> 1;    // 0..3
    const int half   = lane >> 4, l16 = lane & 15;

    // Per-row metadata; row = (2*ipair + p)*16 + r + 8*half
    float sq_i[2][8]; int t_i[2][8];
    #pragma unroll
    for (int p = 0; p < 2; ++p)
        #pragma unroll
        for (int r = 0; r < 8; ++r) {
            const int row = (2 * ipair + p) * 16 + r + 8 * half;
            sq_i[p][r] = sqA[row]; t_i[p][r] = tA[row];
        }

    // Running hard-example stats as squared clipped distances (sqrt deferred: monotone).
    float ap2[2][8], an2[2][8];
    #pragma unroll
    for (int p = 0; p < 2; ++p)
        #pragma unroll
        for (int r = 0; r < 8; ++r) { ap2[p][r] = -FLT_MAX; an2[p][r] = FLT_MAX; }

    const float* Arow0 = As + ((2 * ipair) * 16 + l16) * LDA + 2 * half;
    const float* Brw0  = Bsb + (jgroup * 32 + l16) * LDB + 2 * half;

    int t = 0;
    for (int j0 = 0; j0 < N; j0 += BN) {
        v8f acc[2][2] = {{{}, {}}, {{}, {}}};   // [i-tile p][j-subtile q]

        for (int c = 0; c < NKC; ++c, ++t) {
            // Prefetch next chunk into the other buffer, then wait for current.
            if (wave == 0) {
                const int tn = t + 1;
                if (tn < TCH) {
                    const int jn = (tn / NKC) * BN;
                    const int kn = (tn % NKC) * KC;
                    tdm_load_2d(bsOff + (unsigned)(tn & 1) * bufBytes,
                                X + (size_t)jn * D + kn, KC, BN, D, 6u, 3u);
                    __builtin_amdgcn_s_wait_tensorcnt(1);   // chunk t (and A) done
                } else {
                    __builtin_amdgcn_s_wait_tensorcnt(0);
                }
            }
            __syncthreads();   // buf[t&1] (and As) ready for all waves

            const int kc = c * KC;
            const float* ar0 = Arow0 + kc + (kc >= 256 ? 2 : 0);  // TDM mid-row pad
            const float* ar1 = ar0 + 16 * LDA;
            const float* br0 = Brw0 + (t & 1) * (BN * LDB);
            const float* br1 = br0 + 16 * LDB;

            #pragma unroll 2
            for (int k = 0; k < KC; k += 4) {
                v2f a0 = *(const v2f*)(ar0 + k);
                v2f a1 = *(const v2f*)(ar1 + k);
                v2f b0 = *(const v2f*)(br0 + k);
                v2f b1 = *(const v2f*)(br1 + k);
                acc[0][0] = __builtin_amdgcn_wmma_f32_16x16x4_f32(false, a0, false, b0,
                                                                  (short)0, acc[0][0], false, false);
                acc[0][1] = __builtin_amdgcn_wmma_f32_16x16x4_f32(false, a0, false, b1,
                                                                  (short)0, acc[0][1], false, false);
                acc[1][0] = __builtin_amdgcn_wmma_f32_16x16x4_f32(false, a1, false, b0,
                                                                  (short)0, acc[1][0], false, false);
                acc[1][1] = __builtin_amdgcn_wmma_f32_16x16x4_f32(false, a1, false, b1,
                                                                  (short)0, acc[1][1], false, false);
            }
            __syncthreads();   // all readers of buf[t&1] done -> it may be DMA'd next
        }

        // Epilogue: squared distance, clip, masked running max/min (no sqrt here).
        const int   jb  = j0 + jgroup * 32;
        const int   tj0 = tgt[jb + l16];
        const int   tj1 = tgt[jb + 16 + l16];
        const float sj0 = row_sq[jb + l16];
        const float sj1 = row_sq[jb + 16 + l16];
        #pragma unroll
        for (int p = 0; p < 2; ++p)
            #pragma unroll
            for (int r = 0; r < 8; ++r) {
                const float si = sq_i[p][r]; const int ti = t_i[p][r];
                float d0 = fmaxf(si + sj0 - 2.0f * acc[p][0][r], EPS);
                if (ti == tj0) ap2[p][r] = fmaxf(ap2[p][r], d0);
                else           an2[p][r] = fminf(an2[p][r], d0);
                float d1 = fmaxf(si + sj1 - 2.0f * acc[p][1][r], EPS);
                if (ti == tj1) ap2[p][r] = fmaxf(ap2[p][r], d1);
                else           an2[p][r] = fminf(an2[p][r], d1);
            }
    }

    // Reduce over the 16 N-slots inside each lane half.
    #pragma unroll
    for (int p = 0; p < 2; ++p)
        #pragma unroll
        for (int r = 0; r < 8; ++r) {
            #pragma unroll
            for (int m = 8; m; m >>= 1) {
                ap2[p][r] = fmaxf(ap2[p][r], __shfl_xor(ap2[p][r], m, 16));
                an2[p][r] = fminf(an2[p][r], __shfl_xor(an2[p][r], m, 16));
            }
        }
    __syncthreads();
    if (l16 == 0) {
        #pragma unroll
        for (int p = 0; p < 2; ++p)
            #pragma unroll
            for (int r = 0; r < 8; ++r) {
                const int row = (2 * ipair + p) * 16 + r + 8 * half;
                apJ[jgroup * BM + row] = ap2[p][r];
                anJ[jgroup * BM + row] = an2[p][r];
            }
    }
    __syncthreads();
    if (tid < BM) {
        float a2 = apJ[tid], b2 = anJ[tid];
        #pragma unroll
        for (int g = 1; g < 4; ++g) {
            a2 = fmaxf(a2, apJ[g * BM + tid]);
            b2 = fminf(b2, anJ[g * BM + tid]);
        }
        red[tid] = fmaxf(sqrtf(a2) - sqrtf(b2) + MARGIN, 0.0f);
    }
    __syncthreads();
    if (tid == 0) {
        float s = 0.f;
        for (int tt = 0; tt < BM; ++tt) s += red[tt];   // fixed order -> deterministic
        partial[blockIdx.x] = s;
    }
}

// ---------------------------------------------------------------- cross entropy per row
__global__ void xent_kernel(const float* __restrict__ cls, const int* __restrict__ tgt,
                            float* __restrict__ xrow) {
    __shared__ float sred[256];
    const int row = blockIdx.x;
    const float* cr = cls + (size_t)row * NCLS;
    float v[4]; float mx = -FLT_MAX;
    #pragma unroll
    for (int q = 0; q < 4; ++q) { v[q] = cr[threadIdx.x + q * 256]; mx = fmaxf(mx, v[q]); }
    sred[threadIdx.x] = mx; __syncthreads();
    for (int s = 128; s; s >>= 1) {
        if (threadIdx.x < s) sred[threadIdx.x] = fmaxf(sred[threadIdx.x], sred[threadIdx.x + s]);
        __syncthreads();
    }
    mx = sred[0]; __syncthreads();
    float e = 0.f;
    #pragma unroll
    for (int q = 0; q < 4; ++q) e += expf(v[q] - mx);
    sred[threadIdx.x] = e; __syncthreads();
    for (int s = 128; s; s >>= 1) {
        if (threadIdx.x < s) sred[threadIdx.x] += sred[threadIdx.x + s];
        __syncthreads();
    }
    if (threadIdx.x == 0) {
        const float lse = mx + logf(sred[0]);
        xrow[row] = lse - cr[tgt[row]];
    }
}

// ---------------------------------------------------------------- Frobenius partials
__global__ void frob_kernel(const float* __restrict__ l2, const float* __restrict__ l3,
                            const float* __restrict__ l4, float* __restrict__ part) {
    __shared__ float s42[256], s43[256];
    float a42 = 0.f, a43 = 0.f;
    const int total = N * D;
    for (int i = blockIdx.x * 256 + threadIdx.x; i < total; i += FROB_BLOCKS * 256) {
        const float x4 = l4[i];
        const float d2 = x4 - l2[i]; a42 += d2 * d2;
        const float d3 = x4 - l3[i]; a43 += d3 * d3;
    }
    s42[threadIdx.x] = a42; s43[threadIdx.x] = a43; __syncthreads();
    for (int s = 128; s; s >>= 1) {
        if (threadIdx.x < s) {
            s42[threadIdx.x] += s42[threadIdx.x + s];
            s43[threadIdx.x] += s43[threadIdx.x + s];
        }
        __syncthreads();
    }
    if (threadIdx.x == 0) {
        part[blockIdx.x] = s42[0];
        part[FROB_BLOCKS + blockIdx.x] = s43[0];
    }
}

// ---------------------------------------------------------------- final combine
__global__ void final_kernel(const float* __restrict__ trip_part, const float* __restrict__ xrow,
                             const float* __restrict__ frob_part, float* __restrict__ out) {
    if (threadIdx.x == 0) {
        float ts = 0.f; for (int i = 0; i < N / BM; ++i) ts += trip_part[i];
        float xs = 0.f; for (int i = 0; i < N; ++i) xs += xrow[i];
        float f42 = 0.f, f43 = 0.f;
        for (int i = 0; i < FROB_BLOCKS; ++i) { f42 += frob_part[i]; f43 += frob_part[FROB_BLOCKS + i]; }
        out[0] = ALPHA * (ts / (float)N) + GAMMA * (xs / (float)N)
               + THETA * (sqrtf(f42) + sqrtf(f43));
    }
}

// ---------------------------------------------------------------- launcher
extern "C" void kernel_launch(void* const* d_in, const int* in_sizes, int n_in,
                              void* d_out, int out_size, void* d_ws, size_t ws_size,
                              hipStream_t stream) {
    const float* cls = (const float*)d_in[0];   // [N, NCLS]
    const float* l2  = (const float*)d_in[1];   // [N, D]
    const float* l3  = (const float*)d_in[2];   // [N, D]
    const float* l4  = (const float*)d_in[3];   // [N, D]
    const float* X   = (const float*)d_in[4];   // [N, D] input_fea
    const int*   tgt = (const int*)d_in[5];     // [N]

    float* ws        = (float*)d_ws;
    float* row_sq    = ws;                      // N
    float* trip_part = row_sq + N;              // N/BM
    float* xrow      = trip_part + (N / BM);    // N
    float* frob_part = xrow + N;                // 2*FROB_BLOCKS

    rowsq_kernel<<<N / 8, 256, 0, stream>>>(X, row_sq);

    const size_t shbytes =
        (size_t)(BM * LDA + 2 * BN * LDB + BM /*sqA*/ + BM /*tA*/ + 8 * BM /*apJ,anJ*/ + BM /*red*/)
        * sizeof(float);
    tripdist_kernel<<<N / BM, 256, shbytes, stream>>>(X, tgt, row_sq, trip_part);

    xent_kernel<<<N, 256, 0, stream>>>(cls, tgt, xrow);
    frob_kernel<<<FROB_BLOCKS, 256, 0, stream>>>(l2, l3, l4, frob_part);
    final_kernel<<<1, 32, 0, stream>>>(trip_part, xrow, frob_part, (float*)d_out);
}